// GATv2Layer_35923106464016
// MI455X (gfx1250) — compile-verified
//
#include <hip/hip_runtime.h>
#include <hip/hip_bf16.h>
#include <math.h>

// Problem constants (from reference setup_inputs)
#define BDIM 2
#define NNODES 207
#define TDIM 12
#define FIN 64
#define HEADS 8
#define CDIM 8
#define HC (HEADS * CDIM)       // 64
#define MROWS (BDIM * NNODES * TDIM)  // 4968 rows of X / h
#define LEAKY 0.2f

typedef __attribute__((ext_vector_type(2))) float v2f;
typedef __attribute__((ext_vector_type(8))) float v8f;

// ---------------------------------------------------------------------------
// Kernel 1: h = X @ W  via V_WMMA_F32_16X16X4_F32.
// X: (MROWS, 64) row-major, W: (64, 64) row-major, Hout: (MROWS, 64).
// One wave (32 threads) per 16x16 output tile; 16 WMMA steps over K=64.
// All A/B fragments are preloaded (one load clause, one wait), then the 16
// WMMAs issue back-to-back as two independent accumulation chains.
//
// Per-lane fragment layouts (CDNA5 ISA 7.12.2, 32-bit data):
//  A (16x4):  lanes 0-15: M=lane,    v0=K0, v1=K1 ; lanes 16-31: M=lane-16, v0=K2, v1=K3
//  B (4x16):  lanes 0-15: N=lane,    v0=K0, v1=K1 ; lanes 16-31: N=lane-16, v0=K2, v1=K3
//  C/D (16x16): lane half selects M offset (0/8), vgpr v = M%8, N = lane%16
// ---------------------------------------------------------------------------
__global__ __launch_bounds__(32)
void gatv2_gemm_h(const float* __restrict__ X,
                  const float* __restrict__ W,
                  float* __restrict__ Hout)
{
    const int mtile = blockIdx.x;          // 0..310
    const int ntile = blockIdx.y;          // 0..3
    const int lane  = threadIdx.x;         // 0..31
    const int hi    = lane >> 4;           // 0 = lanes 0-15, 1 = lanes 16-31
    const int l16   = lane & 15;

    const int row0  = mtile * 16 + l16;    // A-matrix row this lane feeds
    const bool rok  = row0 < MROWS;
    const int row   = rok ? row0 : (MROWS - 1);  // clamped: always in-bounds
    const int col   = ntile * 16 + l16;    // B/C/D column this lane holds

    // ---- Preload all fragments (pure load clause, no branches) ----
    v2f afr[16];
    v2f bfr[16];
    const float* __restrict__ Xrow = X + (size_t)row * FIN + hi * 2;
    const float* __restrict__ Wcol = W + (size_t)(hi * 2) * HC + col;

#pragma unroll
    for (int j = 0; j < 16; ++j) {
        const float2 t = *(const float2*)(Xrow + j * 4);   // b64 load
        afr[j].x = t.x;
        afr[j].y = t.y;
    }
#pragma unroll
    for (int j = 0; j < 16; ++j) {
        bfr[j].x = Wcol[(size_t)(j * 4) * HC];
        bfr[j].y = Wcol[(size_t)(j * 4 + 1) * HC];
    }

    // Zero out-of-range rows with value selects (v_cndmask), not exec masks.
#pragma unroll
    for (int j = 0; j < 16; ++j) {
        afr[j].x = rok ? afr[j].x : 0.0f;
        afr[j].y = rok ? afr[j].y : 0.0f;
    }

    // ---- 16 WMMAs, two independent accumulation chains ----
    v8f acc0 = {};
    v8f acc1 = {};
#pragma unroll
    for (int j = 0; j < 16; j += 2) {
        acc0 = __builtin_amdgcn_wmma_f32_16x16x4_f32(
            false, afr[j],     false, bfr[j],     (short)0, acc0, false, false);
        acc1 = __builtin_amdgcn_wmma_f32_16x16x4_f32(
            false, afr[j + 1], false, bfr[j + 1], (short)0, acc1, false, false);
    }
#pragma unroll
    for (int v = 0; v < 8; ++v) acc0[v] += acc1[v];

#pragma unroll
    for (int v = 0; v < 8; ++v) {
        const int r = mtile * 16 + v + hi * 8;
        if (r < MROWS) Hout[(size_t)r * HC + col] = acc0[v];
    }
}

// ---------------------------------------------------------------------------
// Kernel 2: masked-softmax diagonal + scale, fused.
// One block per (b, t, head): 2*12*8 = 192 blocks, 256 threads = 8 waves.
// LDS-stage h[:, head*8 : head*8+8] (207 x 8 f32 = 6.6 KB), then each wave
// processes rows n = wave, wave+8, ... with lanes covering m in 7 chunks of 32.
// ---------------------------------------------------------------------------
__global__ __launch_bounds__(256)
void gatv2_attn_diag(const float* __restrict__ Hbuf,
                     const float* __restrict__ Ahat,
                     const float* __restrict__ avec,
                     float* __restrict__ out)
{
    const int blk  = blockIdx.x;
    const int head = blk & 7;
    const int t    = (blk >> 3) % TDIM;
    const int b    = blk / (TDIM * HEADS);

    __shared__ float hl[NNODES][CDIM];   // h slice for this (b,t,head)
    __shared__ float as[CDIM];

    const int tid = threadIdx.x;
    if (tid < CDIM) as[tid] = avec[tid];

    // Cooperative load: 207 rows x 2 float4 (rows are 16B-aligned in Hbuf).
    for (int idx = tid; idx < NNODES * 2; idx += 256) {
        const int n    = idx >> 1;
        const int part = idx & 1;
        const size_t src =
            (((size_t)b * NNODES + n) * TDIM + t) * HC + head * CDIM + part * 4;
        *(float4*)&hl[n][part * 4] = *(const float4*)&Hbuf[src];
    }
    __syncthreads();

    const int wave = tid >> 5;
    const int lane = tid & 31;

    for (int n = wave; n < NNODES; n += 8) {
        float hn[CDIM];
#pragma unroll
        for (int c = 0; c < CDIM; ++c) hn[c] = hl[n][c];

        const float* __restrict__ Arow = &Ahat[(size_t)n * NNODES];

        // Single pass: keep all 7 lane-chunks of scores in registers.
        float sv[7];
#pragma unroll
        for (int j = 0; j < 7; ++j) {
            const int m = j * 32 + lane;
            float s = -INFINITY;
            if (m < NNODES && Arow[m] > 0.0f) {
                float acc = 0.0f;
#pragma unroll
                for (int c = 0; c < CDIM; ++c) {
                    const float x = hn[c] + hl[m][c];
                    acc = fmaf(as[c], fmaxf(x, LEAKY * x), acc);  // leaky_relu, slope<1
                }
                s = acc;
            }
            sv[j] = s;
        }

        // Row max across 7 chunks then across 32 lanes.
        float mx = sv[0];
#pragma unroll
        for (int j = 1; j < 7; ++j) mx = fmaxf(mx, sv[j]);
#pragma unroll
        for (int off = 16; off > 0; off >>= 1)
            mx = fmaxf(mx, __shfl_xor(mx, off, 32));

        // Row sum of exp(s - max).
        float sum = 0.0f;
#pragma unroll
        for (int j = 0; j < 7; ++j) sum += __expf(sv[j] - mx);
#pragma unroll
        for (int off = 16; off > 0; off >>= 1)
            sum += __shfl_xor(sum, off, 32);

        // Diagonal score e[n,n] (same formula, m == n).
        float enn = -INFINITY;
        if (Arow[n] > 0.0f) {
            float acc = 0.0f;
#pragma unroll
            for (int c = 0; c < CDIM; ++c) {
                const float x = 2.0f * hn[c];
                acc = fmaf(as[c], fmaxf(x, LEAKY * x), acc);
            }
            enn = acc;
        }
        const float diag = __expf(enn - mx) / sum;

        // out layout (B, N, T, H*C) == h's row-major layout: direct store.
        if (lane < CDIM) {
            const size_t dst =
                (((size_t)b * NNODES + n) * TDIM + t) * HC + head * CDIM + lane;
            out[dst] = diag * hn[lane];
        }
    }
}

// ---------------------------------------------------------------------------
extern "C" void kernel_launch(void* const* d_in, const int* in_sizes, int n_in,
                              void* d_out, int out_size, void* d_ws, size_t ws_size,
                              hipStream_t stream)
{
    const float* X    = (const float*)d_in[0];  // (B, N, T, F_IN)
    const float* Ahat = (const float*)d_in[1];  // (N, N)
    const float* W    = (const float*)d_in[2];  // (F_IN, H*C)
    const float* avec = (const float*)d_in[3];  // (C,)
    float* out  = (float*)d_out;                // (B, N, T, H*C)
    float* hbuf = (float*)d_ws;                 // (MROWS, 64) f32 = 1.27 MB

    dim3 g1((MROWS + 15) / 16, HC / 16);        // 311 x 4 tiles
    gatv2_gemm_h<<<g1, 32, 0, stream>>>(X, W, hbuf);

    gatv2_attn_diag<<<BDIM * TDIM * HEADS, 256, 0, stream>>>(hbuf, Ahat, avec, out);
}